// TorchTest_54099408060638
// MI455X (gfx1250) — compile-verified
//
#include <hip/hip_runtime.h>

// ---------------------------------------------------------------------------
// S4 DPLR kernel generation for MI455X (gfx1250, wave32, WMMA)
//   Kernel 1: Cauchy reduction + Woodbury -> pair-interleaved DFT coeffs A_P
//             A_P[l][h] = ( wgt*Re(kf), -wgt*Im(kf) )   (float2, h-contiguous)
//   Kernel 2: irfft as real GEMM (512 x 4098 x 4096) on V_WMMA_F32_16X16X4_F32
//             with double-buffered GLOBAL_LOAD_ASYNC_TO_LDS staging (ASYNCcnt)
// ---------------------------------------------------------------------------

#define Hn   512
#define Nn   32
#define Ln   4096
#define Lf   2049          // L/2 + 1
#define KP   2064          // frequency-bin pairs, 2049 padded to 43*48
#define KCP  48            // pairs staged per LDS chunk (= 96 K rows)
#define NCHUNK 43          // KP / KCP
#define PSTRIDE 144        // LDS row stride in float2: 2*144 mod 64 == 32
                           // -> the two lane halves of ds_load_b64 hit
                           //    disjoint 32-bank groups (conflict-free)

typedef float v2f __attribute__((ext_vector_type(2)));
typedef float v8f __attribute__((ext_vector_type(8)));

// -------------------------- Kernel 1: Cauchy ------------------------------
// grid (2, 2049), block 256: h = fast thread index -> coalesced float2 writes
__global__ __launch_bounds__(256) void s4_cauchy_kernel(
    const float* __restrict__ w_re, const float* __restrict__ w_im,
    const float* __restrict__ p_re, const float* __restrict__ p_im,
    const float* __restrict__ B_re, const float* __restrict__ B_im,
    const float* __restrict__ C_re, const float* __restrict__ C_im,
    const float* __restrict__ log_dt, float2* __restrict__ A_P)
{
    const int h = blockIdx.x * 256 + threadIdx.x;
    const int l = blockIdx.y;
    if (h >= Hn) return;

    const float dt = __expf(log_dt[h]);

    // omega = exp(-2*pi*i*l/L); v_sin/v_cos consume revolutions (x * 2*pi)
    const float u  = (float)l * (1.0f / (float)Ln);
    const float cw = __builtin_amdgcn_cosf(u);
    const float sw = __builtin_amdgcn_sinf(u);
    const float om_re = cw, om_im = -sw;

    // z = 2*(1-omega)/(1+omega)
    const float nzr = 1.0f - om_re, nzi = -om_im;
    const float dzr = 1.0f + om_re, dzi =  om_im;
    const float invz = 1.0f / (dzr * dzr + dzi * dzi);
    const float z_re = 2.0f * (nzr * dzr + nzi * dzi) * invz;
    const float z_im = 2.0f * (nzi * dzr - nzr * dzi) * invz;

    float r00r = 0.f, r00i = 0.f, r01r = 0.f, r01i = 0.f;
    float r10r = 0.f, r10i = 0.f, r11r = 0.f, r11i = 0.f;

    const int base = h * Nn;
    #pragma unroll 4
    for (int n = 0; n < Nn; ++n) {
        const float wr = w_re[base + n] * dt;
        const float wi = w_im[base + n] * dt;
        const float pr = p_re[base + n], pi = p_im[base + n];
        const float br = B_re[base + n], bi = B_im[base + n];
        const float cr = C_re[base + n], ci = C_im[base + n];

        // v_ab = Bc[a] * Cc[b];  Bc = {B, P}, Cc = {C, conj(P)}
        const float v00r = br * cr - bi * ci, v00i = br * ci + bi * cr;
        const float v01r = br * pr + bi * pi, v01i = -br * pi + bi * pr;
        const float v10r = pr * cr - pi * ci, v10i = pr * ci + pi * cr;
        const float v11r = pr * pr + pi * pi;

        // c1 = 1/(z - w),  c2 = 1/(z - conj(w))
        const float d1r = z_re - wr, d1i = z_im - wi;
        const float i1  = 1.0f / (d1r * d1r + d1i * d1i);
        const float c1r = d1r * i1, c1i = -d1i * i1;
        const float d2r = z_re - wr, d2i = z_im + wi;
        const float i2  = 1.0f / (d2r * d2r + d2i * d2i);
        const float c2r = d2r * i2, c2i = -d2i * i2;

        // r_ab += v_ab*c1 + conj(v_ab)*c2
        r00r += v00r * c1r - v00i * c1i + v00r * c2r + v00i * c2i;
        r00i += v00r * c1i + v00i * c1r + v00r * c2i - v00i * c2r;
        r01r += v01r * c1r - v01i * c1i + v01r * c2r + v01i * c2i;
        r01i += v01r * c1i + v01i * c1r + v01r * c2i - v01i * c2r;
        r10r += v10r * c1r - v10i * c1i + v10r * c2r + v10i * c2i;
        r10i += v10r * c1i + v10i * c1r + v10r * c2i - v10i * c2r;
        r11r += v11r * (c1r + c2r);
        r11i += v11r * (c1i + c2i);
    }

    r00r *= dt; r00i *= dt; r01r *= dt; r01i *= dt;
    r10r *= dt; r10i *= dt; r11r *= dt; r11i *= dt;

    // kf = r00 - r01*r10/(1 + r11)
    const float der = 1.0f + r11r, dei = r11i;
    const float ide = 1.0f / (der * der + dei * dei);
    const float nur = r01r * r10r - r01i * r10i;
    const float nui = r01r * r10i + r01i * r10r;
    const float qr  = (nur * der + nui * dei) * ide;
    const float qi  = (nui * der - nur * dei) * ide;
    const float kr = r00r - qr, ki = r00i - qi;

    // kf *= 2/(1+omega)
    const float idw = 1.0f / (dzr * dzr + dzi * dzi);
    const float fr  = 2.0f * dzr * idw, fi = -2.0f * dzi * idw;
    const float kfr = kr * fr - ki * fi;
    const float kfi = kr * fi + ki * fr;

    // irfft weights: bins 0 and L/2 count once, others twice; 1/L norm.
    const float wgt = (l == 0 || l == Ln / 2) ? (1.0f / (float)Ln)
                                              : (2.0f / (float)Ln);
    A_P[(size_t)l * Hn + h] = make_float2(wgt * kfr, -wgt * kfi);
}

// ---------------------- Kernel 2: irfft via WMMA --------------------------
// grid (L/128, H/128), block 256 (8 waves). Wave w owns t-tile [t0, t0+16),
// accumulates 8 h-tiles; A chunks stream through double-buffered LDS via
// global_load_async_to_lds_b128 (ASYNCcnt), overlapped with WMMA compute.
__global__ __launch_bounds__(256) void s4_irfft_wmma_kernel(
    const float2* __restrict__ A_P, float* __restrict__ out)
{
    __shared__ float2 ldsA[2][KCP * PSTRIDE];   // 2 x 54 KB

    const int tid  = threadIdx.x;
    const int wave = tid >> 5;
    const int lane = tid & 31;
    const int laneHalf = lane >> 4;             // 0 or 1
    const int laneMod  = lane & 15;

    const int t0   = blockIdx.x * 128 + wave * 16;
    const int h0   = blockIdx.y * 128;
    const int tcol = t0 + laneMod;

    // Stage one KCP x 128 float2 tile into LDS buffer `buf` (async, no VGPR
    // data path). 3072 x 16B per block -> 12 ops per thread.
    auto stage = [&](int chunk, int buf) {
        const int kp0 = chunk * KCP;
        #pragma unroll
        for (int j = 0; j < 12; ++j) {
            const int i  = tid + j * 256;       // 0..3071
            const int r  = i >> 6;              // pair-row 0..47
            const int c4 = i & 63;              // float4 within row
            const float2* gptr = A_P + (size_t)(kp0 + r) * Hn + h0 + 2 * c4;
            const unsigned laddr =
                (unsigned)(uintptr_t)&ldsA[buf][r * PSTRIDE + 2 * c4];
            asm volatile("global_load_async_to_lds_b128 %0, %1, off"
                         :: "v"(laddr), "v"(gptr) : "memory");
        }
    };

    v8f acc[8];
    #pragma unroll
    for (int tt = 0; tt < 8; ++tt)
        acc[tt] = (v8f){0.f, 0.f, 0.f, 0.f, 0.f, 0.f, 0.f, 0.f};

    stage(0, 0);
    asm volatile("s_wait_asynccnt 0x0" ::: "memory");
    __syncthreads();

    for (int chunk = 0; chunk < NCHUNK; ++chunk) {
        const int buf = chunk & 1;
        if (chunk + 1 < NCHUNK) stage(chunk + 1, buf ^ 1);

        const int kp0 = chunk * KCP;
        const float2* lbase = &ldsA[buf][0];

        #pragma unroll 4
        for (int s = 0; s < KCP / 2; ++s) {
            const int kp = 2 * s + laneHalf;          // local pair row
            const int l  = kp0 + kp;                  // frequency bin
            // exact phase: 2*pi*((l*t) mod 4096)/4096, fed as revolutions
            const unsigned m = ((unsigned)(l * tcol)) & (unsigned)(Ln - 1);
            const float ang  = (float)m * (1.0f / (float)Ln);
            v2f b;
            b.x = __builtin_amdgcn_cosf(ang);         // K row 2l   (cos)
            b.y = __builtin_amdgcn_sinf(ang);         // K row 2l+1 (sin)

            const float2* lp = lbase + kp * PSTRIDE + laneMod;
            #pragma unroll
            for (int tt = 0; tt < 8; ++tt) {
                const float2 av = lp[16 * tt];        // one ds_load_b64
                v2f a; a.x = av.x; a.y = av.y;        // (A[h][2l], A[h][2l+1])
                acc[tt] = __builtin_amdgcn_wmma_f32_16x16x4_f32(
                    false, a, false, b, (short)0, acc[tt], false, false);
            }
        }

        asm volatile("s_wait_asynccnt 0x0" ::: "memory");
        __syncthreads();
    }

    // D layout: VGPR r holds M = r + 8*(lane>>4), N = lane&15
    #pragma unroll
    for (int tt = 0; tt < 8; ++tt) {
        #pragma unroll
        for (int r = 0; r < 8; ++r) {
            const int hh = h0 + 16 * tt + r + 8 * laneHalf;
            out[hh * Ln + tcol] = acc[tt][r];
        }
    }
}

// ---------------------------------------------------------------------------
extern "C" void kernel_launch(void* const* d_in, const int* in_sizes, int n_in,
                              void* d_out, int out_size, void* d_ws, size_t ws_size,
                              hipStream_t stream) {
    const float* w_re   = (const float*)d_in[0];
    const float* w_im   = (const float*)d_in[1];
    const float* p_re   = (const float*)d_in[2];
    const float* p_im   = (const float*)d_in[3];
    const float* B_re   = (const float*)d_in[4];
    const float* B_im   = (const float*)d_in[5];
    const float* C_re   = (const float*)d_in[6];
    const float* C_im   = (const float*)d_in[7];
    const float* log_dt = (const float*)d_in[8];
    // d_in[9] = L (always 4096 here)

    float2* A_P = (float2*)d_ws;                     // KP x H float2 = 8.45 MB

    // Zero pad pair-rows [2049, KP); graph-capture safe.
    hipMemsetAsync(d_ws, 0, (size_t)KP * Hn * sizeof(float2), stream);

    s4_cauchy_kernel<<<dim3(Hn / 256, Lf), 256, 0, stream>>>(
        w_re, w_im, p_re, p_im, B_re, B_im, C_re, C_im, log_dt, A_P);

    s4_irfft_wmma_kernel<<<dim3(Ln / 128, Hn / 128), 256, 0, stream>>>(
        A_P, (float*)d_out);
}